// Model_25726854103373
// MI455X (gfx1250) — compile-verified
//
#include <hip/hip_runtime.h>

typedef __attribute__((ext_vector_type(2))) float v2f;
typedef __attribute__((ext_vector_type(8))) float v8f;
typedef int v4i __attribute__((vector_size(16)));

#if defined(__has_builtin)
#if __has_builtin(__builtin_amdgcn_global_load_async_to_lds_b128) && \
    __has_builtin(__builtin_amdgcn_s_wait_asynccnt)
#define USE_ASYNC_LDS 1
#endif
#endif
#ifndef USE_ASYNC_LDS
#define USE_ASYNC_LDS 0
#endif

static __device__ __forceinline__ void atomAddF(float* p, float v) {
  unsafeAtomicAdd(p, v);
}

// ---------------------------------------------------------------- utility
__global__ void fill_f32(float* __restrict__ p, int n, float v) {
  int i = blockIdx.x * blockDim.x + threadIdx.x;
  if (i < n) p[i] = v;
}

__global__ void deg_count(const int* __restrict__ src, const int* __restrict__ dst,
                          float* __restrict__ dout, float* __restrict__ din, int E) {
  int e = blockIdx.x * blockDim.x + threadIdx.x;
  if (e >= E) return;
  atomAddF(&dout[src[e]], 1.0f);
  atomAddF(&din[dst[e]], 1.0f);
}

__global__ void inv_sqrt_clip(float* __restrict__ p, int n) {
  int i = blockIdx.x * blockDim.x + threadIdx.x;
  if (i < n) p[i] = rsqrtf(fmaxf(p[i], 1.0f));
}

// c-layer-0 input: [NC x 40] = concat(cfeats[NC x 32], op_emb[ctypes] [NC x 8])
__global__ void build_c0(const float* __restrict__ cf, const int* __restrict__ ct,
                         const float* __restrict__ ope, float* __restrict__ out, int NCn) {
  int i = blockIdx.x * blockDim.x + threadIdx.x;
  if (i >= NCn * 40) return;
  int n = i / 40, c = i - n * 40;
  out[i] = (c < 32) ? cf[n * 32 + c] : ope[ct[n] * 8 + (c - 32)];
}

// agg[dst[e], k4..k4+3] += feat[src[e], k4..k4+3] * (scale ? scale[src[e]] : 1)
__global__ void scatter_add4(const float* __restrict__ feat, int fdim,
                             const int* __restrict__ src, const int* __restrict__ dst,
                             const float* __restrict__ scale,
                             float* __restrict__ agg, int E) {
  int tid = blockIdx.x * blockDim.x + threadIdx.x;
  int perE = fdim >> 2;                    // fdim is always a multiple of 4
  if (tid >= E * perE) return;
  int e = tid / perE;
  int k4 = (tid - e * perE) << 2;
  int s = src[e], d = dst[e];
  float sc = scale ? scale[s] : 1.0f;
  const float4 f = *reinterpret_cast<const float4*>(feat + (size_t)s * fdim + k4);
  float* ap = agg + (size_t)d * fdim + k4;
  atomAddF(ap + 0, f.x * sc);
  atomAddF(ap + 1, f.y * sc);
  atomAddF(ap + 2, f.z * sc);
  atomAddF(ap + 3, f.w * sc);
}

// gbuf[gid[n], k] += feat[n, k]   (feature width 64)
__global__ void group_scatter(const float* __restrict__ feat, const int* __restrict__ gid,
                              float* __restrict__ gbuf, int n) {
  int i = blockIdx.x * blockDim.x + threadIdx.x;
  if (i >= n * 64) return;
  int node = i >> 6, k = i & 63;
  atomAddF(&gbuf[gid[node] * 64 + k], feat[i]);
}

__global__ void final_nccl(const float* __restrict__ g, const float* __restrict__ w,
                           const float* __restrict__ b, float* __restrict__ out, int ng) {
  int i = blockIdx.x * blockDim.x + threadIdx.x;
  if (i >= ng) return;
  float s = b[0];
#pragma unroll 8
  for (int k = 0; k < 64; ++k) s += g[i * 64 + k] * w[k];
  out[i] = s;
}

__global__ void final_strategy(const float* __restrict__ g, const float* __restrict__ w,
                               const float* __restrict__ b, float* __restrict__ out, int ng) {
  int i = blockIdx.x * blockDim.x + threadIdx.x;
  if (i >= ng * 8) return;
  int gi = i >> 3, j = i & 7;
  float s = b[j];
#pragma unroll 8
  for (int k = 0; k < 64; ++k) s += g[gi * 64 + k] * w[k * 8 + j];
  out[i] = s;
}

// ---------------------------------------------------------------- WMMA GEMM
// Out[m,n] = act( (A@W[0:Kf] + Ef@W[Kf:Kf+edim])[m,n] * s_in[m] + bias[n] )
// A: [M x Kf], Ef: [M x edim], W: [(Kf+edim) x N] row-major.
// Block = 128 threads (4 waves). Each wave: 32(M) x 64(N) tile.
// W K-chunks are staged into double-buffered LDS with async-to-LDS loads
// (ASYNCcnt + s_wait_asynccnt), shared by all 4 waves.
#define KC 32

static __device__ __forceinline__ v8f wmma4(v2f a, v2f b, v8f c) {
  return __builtin_amdgcn_wmma_f32_16x16x4_f32(false, a, false, b, (short)0, c,
                                               false, false);
}

// Compile-time NSTEPS so a0/a1 fully unroll into registers (no v_movrels).
template <int NSTEPS>
static __device__ __forceinline__ void chunk_mma(const float* __restrict__ Ar0,
                                                 const float* __restrict__ Ar1,
                                                 int kk, int koff,
                                                 const float* __restrict__ sb,
                                                 v8f acc0[4], v8f acc1[4]) {
  v2f a0[NSTEPS], a1[NSTEPS];
#pragma unroll
  for (int s = 0; s < NSTEPS; ++s) {
    const float* p0 = Ar0 + kk + 4 * s + koff;
    const float* p1 = Ar1 + kk + 4 * s + koff;
    a0[s].x = p0[0]; a0[s].y = p0[1];
    a1[s].x = p1[0]; a1[s].y = p1[1];
  }
#pragma unroll
  for (int s = 0; s < NSTEPS; ++s) {
#pragma unroll
    for (int j = 0; j < 4; ++j) {
      v2f b;
      b.x = sb[s * 256 + 16 * j];        // K row 4s+koff
      b.y = sb[s * 256 + 64 + 16 * j];   // K row 4s+koff+1
      acc0[j] = wmma4(a0[s], b, acc0[j]);
      acc1[j] = wmma4(a1[s], b, acc1[j]);
    }
  }
}

__global__ void __launch_bounds__(128) gemm_gconv(
    const float* __restrict__ A, int Kf,
    const float* __restrict__ Ef, int edim,
    const float* __restrict__ W, const float* __restrict__ bias,
    const float* __restrict__ s_in,
    float* __restrict__ Out,
    int M, int N, int doRelu) {
  __shared__ float sW[2][KC * 64];

  const int tid   = threadIdx.x;
  const int lane  = tid & 31;
  const int wave  = tid >> 5;
  const int lhalf = lane >> 4;            // 0: K pair {0,1}, 1: K pair {2,3}
  const int lmod  = lane & 15;
  const int koff  = lhalf * 2;
  const int nbase = blockIdx.y * 64;
  const int wtile = blockIdx.x * (blockDim.x >> 5) + wave;  // 32-row tile id
  const int mbase = wtile * 32;
  const bool active = (mbase < M);        // wave-uniform; inactive waves still barrier

  v8f acc0[4] = {};
  v8f acc1[4] = {};

  const int nchunks = (Kf + KC - 1) / KC;

  auto stage = [&](int kk, int csz, int b) {
    const float* Wg = W + (size_t)kk * N + nbase;
    for (int f = tid; f < csz * 16; f += 128) {
      int r  = f >> 4;
      int c4 = (f & 15) << 2;
#if USE_ASYNC_LDS
      __builtin_amdgcn_global_load_async_to_lds_b128(
          (__attribute__((address_space(1))) v4i*)(Wg + (size_t)r * N + c4),
          (__attribute__((address_space(3))) v4i*)(&sW[b][r * 64 + c4]),
          0, 0);
#else
      const float4 v = *reinterpret_cast<const float4*>(Wg + (size_t)r * N + c4);
      *reinterpret_cast<float4*>(&sW[b][r * 64 + c4]) = v;
#endif
    }
  };
  auto waitStage = [&]() {
#if USE_ASYNC_LDS
    __builtin_amdgcn_s_wait_asynccnt(0);
#endif
    __syncthreads();
  };

  stage(0, (Kf < KC) ? Kf : KC, 0);
  waitStage();

  const float* Ar0 = A + (size_t)(mbase + lmod) * Kf;        // wave rows 0..15
  const float* Ar1 = Ar0 + (size_t)16 * Kf;                  // wave rows 16..31

  for (int p = 0; p < nchunks; ++p) {
    const int kk  = p * KC;
    const int csz = (Kf - kk < KC) ? (Kf - kk) : KC;
    if (p + 1 < nchunks) {
      const int nk = kk + KC;
      stage(nk, (Kf - nk < KC) ? (Kf - nk) : KC, (p + 1) & 1);
    }
    if (active) {
      const float* sb = &sW[p & 1][koff * 64 + lmod];
      switch (csz >> 2) {       // only 8 (K=512 chunks), 2 (Kf=40 rem), 4 (Kf=16)
        case 8: chunk_mma<8>(Ar0, Ar1, kk, koff, sb, acc0, acc1); break;
        case 4: chunk_mma<4>(Ar0, Ar1, kk, koff, sb, acc0, acc1); break;
        case 2: chunk_mma<2>(Ar0, Ar1, kk, koff, sb, acc0, acc1); break;
        case 6: chunk_mma<6>(Ar0, Ar1, kk, koff, sb, acc0, acc1); break;
        case 1: chunk_mma<1>(Ar0, Ar1, kk, koff, sb, acc0, acc1); break;
        default: break;
      }
    }
    waitStage();
  }

  if (!active) return;

  // edge-feature K segment (agg_e part of the concat), <= 2 k-steps: direct
  const float* Er0 = Ef + (size_t)(mbase + lmod) * edim;
  const float* Er1 = Er0 + (size_t)16 * edim;
  for (int kk = 0; kk < edim; kk += 4) {
    v2f a0, a1;
    a0.x = Er0[kk + koff]; a0.y = Er0[kk + koff + 1];
    a1.x = Er1[kk + koff]; a1.y = Er1[kk + koff + 1];
    const float* Wr0 = W + (size_t)(Kf + kk + koff) * N + nbase + lmod;
    const float* Wr1 = Wr0 + N;
#pragma unroll
    for (int j = 0; j < 4; ++j) {
      v2f b;
      b.x = Wr0[16 * j];
      b.y = Wr1[16 * j];
      acc0[j] = wmma4(a0, b, acc0[j]);
      acc1[j] = wmma4(a1, b, acc1[j]);
    }
  }

  // epilogue: C/D layout -> n = nbase+16j+(lane&15), m = mbase + r + 8*lhalf (+16)
#pragma unroll
  for (int j = 0; j < 4; ++j) {
    int n = nbase + 16 * j + lmod;
    float bn = bias[n];
#pragma unroll
    for (int r = 0; r < 8; ++r) {
      int m0 = mbase + r + lhalf * 8;
      int m1 = m0 + 16;
      float v0 = acc0[j][r] * s_in[m0] + bn;
      float v1 = acc1[j][r] * s_in[m1] + bn;
      if (doRelu) { v0 = fmaxf(v0, 0.0f); v1 = fmaxf(v1, 0.0f); }
      Out[(size_t)m0 * N + n] = v0;
      Out[(size_t)m1 * N + n] = v1;
    }
  }
}

// ---------------------------------------------------------------- host side
extern "C" void kernel_launch(void* const* d_in, const int* in_sizes, int n_in,
                              void* d_out, int out_size, void* d_ws, size_t ws_size,
                              hipStream_t stream) {
  constexpr int NCn = 20000, NTn = 20000, ECn = 320000, ETn = 320000, NGn = 2000;
  constexpr int NH = 512;

  const float* cfeats = (const float*)d_in[0];
  const float* cedge  = (const float*)d_in[1];
  const int*   ctypes = (const int*)d_in[2];
  const float* tfeats = (const float*)d_in[3];
  const float* tedge  = (const float*)d_in[4];
  const int*   c_src  = (const int*)d_in[5];
  const int*   c_dst  = (const int*)d_in[6];
  const int*   t_src  = (const int*)d_in[7];
  const int*   t_dst  = (const int*)d_in[8];
  const int*   cgid   = (const int*)d_in[9];
  const int*   tgid   = (const int*)d_in[10];
  const float* op_emb = (const float*)d_in[11];
  const float* cW[5], *cb[5], *tW[5], *tb[5];
  for (int i = 0; i < 5; ++i) { cW[i] = (const float*)d_in[12 + i]; cb[i] = (const float*)d_in[17 + i]; }
  for (int i = 0; i < 5; ++i) { tW[i] = (const float*)d_in[22 + i]; tb[i] = (const float*)d_in[27 + i]; }
  const float* fs_W = (const float*)d_in[32];
  const float* fs_b = (const float*)d_in[33];
  const float* fn_W = (const float*)d_in[34];
  const float* fn_b = (const float*)d_in[35];
  float* out = (float*)d_out;

  // workspace carve-up
  char* ws = (char*)d_ws;
  size_t off = 0;
  auto carve = [&](size_t bytes) -> float* {
    float* p = (float*)(ws + off);
    off = (off + bytes + 255) & ~(size_t)255;
    return p;
  };
  float* bufF0  = carve((size_t)NCn * NH * 4);
  float* bufF1  = carve((size_t)NCn * NH * 4);
  float* bufAgg = carve((size_t)NCn * NH * 4);
  float* s_cout = carve(NCn * 4);
  float* s_cin  = carve(NCn * 4);
  float* s_tout = carve(NTn * 4);
  float* s_tin  = carve(NTn * 4);
  float* aggEc  = carve((size_t)NCn * 8 * 4);
  float* aggEt  = carve((size_t)NTn * 4 * 4);
  float* gbuf   = carve((size_t)NGn * 64 * 4);

  auto cdiv = [](long a, long b) { return (int)((a + b - 1) / b); };
  const int TB = 256;

  // ---- loop-invariant precompute: degrees, deg^-0.5, edge-feature agg -----
  fill_f32<<<cdiv(NCn, TB), TB, 0, stream>>>(s_cout, NCn, 0.0f);
  fill_f32<<<cdiv(NCn, TB), TB, 0, stream>>>(s_cin, NCn, 0.0f);
  fill_f32<<<cdiv(NTn, TB), TB, 0, stream>>>(s_tout, NTn, 0.0f);
  fill_f32<<<cdiv(NTn, TB), TB, 0, stream>>>(s_tin, NTn, 0.0f);
  deg_count<<<cdiv(ECn, TB), TB, 0, stream>>>(c_src, c_dst, s_cout, s_cin, ECn);
  deg_count<<<cdiv(ETn, TB), TB, 0, stream>>>(t_src, t_dst, s_tout, s_tin, ETn);
  inv_sqrt_clip<<<cdiv(NCn, TB), TB, 0, stream>>>(s_cout, NCn);
  inv_sqrt_clip<<<cdiv(NCn, TB), TB, 0, stream>>>(s_cin, NCn);
  inv_sqrt_clip<<<cdiv(NTn, TB), TB, 0, stream>>>(s_tout, NTn);
  inv_sqrt_clip<<<cdiv(NTn, TB), TB, 0, stream>>>(s_tin, NTn);

  fill_f32<<<cdiv(NCn * 8, TB), TB, 0, stream>>>(aggEc, NCn * 8, 0.0f);
  fill_f32<<<cdiv(NTn * 4, TB), TB, 0, stream>>>(aggEt, NTn * 4, 0.0f);
  scatter_add4<<<cdiv((long)ECn * 2, TB), TB, 0, stream>>>(cedge, 8, c_src, c_dst, nullptr, aggEc, ECn);
  scatter_add4<<<cdiv((long)ETn * 1, TB), TB, 0, stream>>>(tedge, 4, t_src, t_dst, nullptr, aggEt, ETn);

  const int wtiles = NCn / 32;               // 625 (also NT/32)
  const int gemmBX = cdiv(wtiles, 4);        // 4 waves / 128-thread block

  // ------------------------------- c path ---------------------------------
  build_c0<<<cdiv(NCn * 40, TB), TB, 0, stream>>>(cfeats, ctypes, op_emb, bufF0, NCn);
  {
    const int inw[5]  = {40, NH, NH, NH, NH};
    const int outw[5] = {NH, NH, NH, NH, 64};
    const float* cur = bufF0;
    float* nxt = bufF1;
    for (int i = 0; i < 5; ++i) {
      int agg_n = NCn * inw[i];
      fill_f32<<<cdiv(agg_n, TB), TB, 0, stream>>>(bufAgg, agg_n, 0.0f);
      scatter_add4<<<cdiv((long)ECn * (inw[i] / 4), TB), TB, 0, stream>>>(
          cur, inw[i], c_src, c_dst, s_cout, bufAgg, ECn);
      gemm_gconv<<<dim3(gemmBX, outw[i] / 64), 128, 0, stream>>>(
          bufAgg, inw[i], aggEc, 8, cW[i], cb[i], s_cin, nxt, NCn, outw[i], i < 4);
      const float* tmp = cur; cur = nxt; nxt = (float*)tmp;
      if (i == 0) nxt = bufF0;               // after L0: cur=bufF1, nxt=bufF0
    }
    // cur = final c embedding [NC x 64]
    fill_f32<<<cdiv(NGn * 64, TB), TB, 0, stream>>>(gbuf, NGn * 64, 0.0f);
    group_scatter<<<cdiv(NCn * 64, TB), TB, 0, stream>>>(cur, cgid, gbuf, NCn);
    final_nccl<<<cdiv(NGn, TB), TB, 0, stream>>>(gbuf, fn_W, fn_b, out, NGn);
  }

  // ------------------------------- t path ---------------------------------
  {
    const int inw[5]  = {16, NH, NH, NH, NH};
    const int outw[5] = {NH, NH, NH, NH, 64};
    const float* cur = tfeats;
    float* nxt = bufF0;
    for (int i = 0; i < 5; ++i) {
      int agg_n = NTn * inw[i];
      fill_f32<<<cdiv(agg_n, TB), TB, 0, stream>>>(bufAgg, agg_n, 0.0f);
      scatter_add4<<<cdiv((long)ETn * (inw[i] / 4), TB), TB, 0, stream>>>(
          cur, inw[i], t_src, t_dst, s_tout, bufAgg, ETn);
      gemm_gconv<<<dim3(gemmBX, outw[i] / 64), 128, 0, stream>>>(
          bufAgg, inw[i], aggEt, 4, tW[i], tb[i], s_tin, nxt, NTn, outw[i], i < 4);
      cur = nxt;
      nxt = (cur == bufF0) ? bufF1 : bufF0;
    }
    // cur = final t embedding [NT x 64]
    fill_f32<<<cdiv(NGn * 64, TB), TB, 0, stream>>>(gbuf, NGn * 64, 0.0f);
    group_scatter<<<cdiv(NTn * 64, TB), TB, 0, stream>>>(cur, tgid, gbuf, NTn);
    final_strategy<<<cdiv(NGn * 8, TB), TB, 0, stream>>>(gbuf, fs_W, fs_b, out + NGn, NGn);
  }
}